// PhotonicLayer_86870008528899
// MI455X (gfx1250) — compile-verified
//
#include <hip/hip_runtime.h>

// CDNA5 / gfx1250: wave32, WMMA f32 16x16x4, Tensor Data Mover.
typedef __attribute__((ext_vector_type(2))) float v2f;
typedef __attribute__((ext_vector_type(4))) float v4f;
typedef __attribute__((ext_vector_type(8))) float v8f;
typedef __attribute__((ext_vector_type(4))) unsigned int v4u;
typedef __attribute__((ext_vector_type(8))) unsigned int v8u;

#define LOSS_AMP 0.994260073f   // 10^(-0.05/20) per-MZI amplitude transmission
#define KERR     2.6e-17f       // 2.6e-20 * 1000

#define XS 68                   // padded LDS row stride (floats): conflict-free b64 frags
#define TILE_FLOATS (64 * XS)   // 4352 floats per 64x64 staged tile
#define SMEM_BYTES  (4 * TILE_FLOATS * 4)  // xr, xi, Ur, Ui tiles = 69632 B (dynamic LDS)

// ---------------------------------------------------------------------------
// Kernel 1: compose the 64x64 complex unitary from 2016 MZI rotations.
// One workgroup, 64 threads; thread t owns column t of U. Every rotation only
// mixes rows (m, m+1) within a column, so threads never interact -> no
// barriers needed. LDS layout [row][thread] => bank == thread, conflict-free.
// ---------------------------------------------------------------------------
__global__ void __launch_bounds__(64) build_u_kernel(
    const float* __restrict__ phases,
    float* __restrict__ Ur, float* __restrict__ Ui) {
  __shared__ float cR[64 * 64];
  __shared__ float cI[64 * 64];
  const int t = threadIdx.x;

  for (int r = 0; r < 64; ++r) {
    cR[r * 64 + t] = (r == t) ? 1.0f : 0.0f;
    cI[r * 64 + t] = 0.0f;
  }

  int idx = 0;
  for (int i = 1; i < 64; ++i) {
    for (int j = 0; j < i; ++j) {
      const float th = phases[idx++];           // uniform (scalar) load
      const float cc = LOSS_AMP * cosf(th);
      const float ss = LOSS_AMP * sinf(th);
      const float ar = cR[j * 64 + t];
      const float ai = cI[j * 64 + t];
      const float br = cR[(j + 1) * 64 + t];
      const float bi = cI[(j + 1) * 64 + t];
      // T = loss * [[c, i s],[i s, c]] applied to rows (j, j+1)
      cR[j * 64 + t]       = cc * ar - ss * bi;
      cI[j * 64 + t]       = cc * ai + ss * br;
      cR[(j + 1) * 64 + t] = cc * br - ss * ai;
      cI[(j + 1) * 64 + t] = cc * bi + ss * ar;
    }
  }

  for (int r = 0; r < 64; ++r) {
    Ur[r * 64 + t] = cR[r * 64 + t];
    Ui[r * 64 + t] = cI[r * 64 + t];
  }
}

// ---------------------------------------------------------------------------
// Tensor Data Mover: DMA one 64x64 f32 tile (row stride 64 floats in global)
// into LDS at byte offset lds_off, inserting 4 DWORDs of padding after every
// 64 DWORDs -> LDS row stride 68 floats, matching the WMMA fragment
// addressing. Inline asm (portable across both toolchains; the clang builtin
// arity differs between them). D# bitfields per CDNA5 ISA ch.8:
//   group0: count | lds_addr | global_addr | type=2
//   group1: data_size/pad cfg | tensor dims | tile dims | stride0
//   group2/3: zero (2D tensor)
// Tracked with TENSORcnt.
// ---------------------------------------------------------------------------
__device__ __forceinline__ void tdm_load_tile(const float* gsrc, unsigned lds_off) {
  const unsigned long long ga = (unsigned long long)gsrc;
  const v4u g0 = {
      1u,                                            // count=1, user mode
      lds_off,                                       // lds_addr (bytes)
      (unsigned)(ga & 0xFFFFFFFFu),                  // global_addr[31:0]
      (unsigned)(((ga >> 32) & 0x01FFFFFFu)          // global_addr[56:32]
                 | (2u << 30))};                     // type=2 ("image")
  const v8u g1 = {
      (2u << 16)            // data_size = 4 bytes
          | (1u << 20)      // pad_enable
          | (5u << 22)      // pad_interval code 5 = every 64 DWORDs
          | (3u << 25),     // pad_amount  code 3 = 4 DWORDs
      64u << 16,            // tensor_dim0 = 64 (bits 79:48)
      64u << 16,            // tensor_dim1 = 64 (bits 111:80)
      64u << 16,            // tile_dim0   = 64 (bits 127:112)
      64u,                  // tile_dim1   = 64 (bits 143:128), tile_dim2 = 0
      64u,                  // tensor_dim0_stride = 64 (bits 207:160)
      0u, 0u};              // tensor_dim1_stride = 0 (unused for 2D)
  const v4u gz = {0u, 0u, 0u, 0u};
  asm volatile("tensor_load_to_lds %0, %1, %2, %3"
               :
               : "s"(g0), "s"(g1), "s"(gz), "s"(gz)
               : "memory");
}

// ---------------------------------------------------------------------------
// Kernel 2: out = (xr + i*xi) @ U^T, then Kerr phase rotation.
//   out_r = xr@Ur^T - xi@Ui^T ; out_i = xr@Ui^T + xi@Ur^T
// 128 threads = 4 waves; each wave owns a 16-row strip of a 64-row tile.
// x tile staged into padded LDS by the Tensor Data Mover (wave 0 issues the
// DMA; TDM ignores EXEC so other waves branch around it). U (32 KB, L2-hot)
// staged with coalesced loads that overlap the DMA.
// WMMA A-frag (16x4 f32): lane<16 holds row=lane, K={4k,4k+1}; lane>=16 holds
// K={4k+2,4k+3} -> one aligned float2 read. B-frag (4x16, B[k][j]=U[j][k])
// has the identical per-lane pattern. Stride 68 => banks 4*lane+{0..3},
// conflict-free for both (compiler fuses them into ds_load_2addr_b64).
// ---------------------------------------------------------------------------
__global__ void __launch_bounds__(128) photonic_kernel(
    const float* __restrict__ xr, const float* __restrict__ xi,
    const float* __restrict__ Ur, const float* __restrict__ Ui,
    float* __restrict__ out) {
  extern __shared__ float smem[];
  float* lXr = smem;
  float* lXi = smem + TILE_FLOATS;
  float* lUr = smem + 2 * TILE_FLOATS;
  float* lUi = smem + 3 * TILE_FLOATS;

  const int tid   = threadIdx.x;
  const long base = (long)blockIdx.x * (64 * 64);  // first element of x tile

  if (tid < 32) {  // wave 0 issues the DMA for the streamed x tiles
    tdm_load_tile(xr + base, (unsigned)(unsigned long long)lXr);
    tdm_load_tile(xi + base, (unsigned)(unsigned long long)lXi);
  }

  // Stage U with coalesced loads (overlaps the TDM transfer above).
  for (int q = tid; q < 1024; q += 128) {
    const int r = q >> 4;
    const int c = (q & 15) << 2;
    const v4f vr = *((const v4f*)Ur + q);
    const v4f vi = *((const v4f*)Ui + q);
    *(v4f*)&lUr[r * XS + c] = vr;
    *(v4f*)&lUi[r * XS + c] = vi;
  }

  if (tid < 32) __builtin_amdgcn_s_wait_tensorcnt(0);  // drain wave 0's DMA
  __syncthreads();                                     // publish LDS to block

  const int wave  = tid >> 5;
  const int lane  = tid & 31;
  const int lmod  = lane & 15;
  const int lhalf = lane >> 4;
  const int arow  = wave * 16 + lmod;       // A-matrix row within tile
  const long brow = (long)blockIdx.x * 64;  // batch row base

  for (int n = 0; n < 4; ++n) {             // N-tiles of 16 output columns
    v8f accA = {0.f, 0.f, 0.f, 0.f, 0.f, 0.f, 0.f, 0.f};  // xr . Ur
    v8f accB = {0.f, 0.f, 0.f, 0.f, 0.f, 0.f, 0.f, 0.f};  // xi . Ui
    v8f accC = {0.f, 0.f, 0.f, 0.f, 0.f, 0.f, 0.f, 0.f};  // xr . Ui
    v8f accD = {0.f, 0.f, 0.f, 0.f, 0.f, 0.f, 0.f, 0.f};  // xi . Ur
    const int j = n * 16 + lmod;            // output column / U row

#pragma unroll
    for (int ks = 0; ks < 16; ++ks) {       // K = 64 in steps of 4
      const int k0 = ks * 4 + lhalf * 2;
      const v2f a_r = *(const v2f*)&lXr[arow * XS + k0];
      const v2f a_i = *(const v2f*)&lXi[arow * XS + k0];
      const v2f b_r = *(const v2f*)&lUr[j * XS + k0];
      const v2f b_i = *(const v2f*)&lUi[j * XS + k0];

      accA = __builtin_amdgcn_wmma_f32_16x16x4_f32(
          false, a_r, false, b_r, (short)0, accA, false, false);
      accB = __builtin_amdgcn_wmma_f32_16x16x4_f32(
          false, a_i, false, b_i, (short)0, accB, false, false);
      accC = __builtin_amdgcn_wmma_f32_16x16x4_f32(
          false, a_r, false, b_i, (short)0, accC, false, false);
      accD = __builtin_amdgcn_wmma_f32_16x16x4_f32(
          false, a_i, false, b_r, (short)0, accD, false, false);
    }

    // Epilogue: combine complex parts, apply Kerr nonlinearity, store NT
    // (output is write-once; keep it out of L2).
#pragma unroll
    for (int v = 0; v < 8; ++v) {
      const float orr = accA[v] - accB[v];
      const float oii = accC[v] + accD[v];
      const float ph  = KERR * (orr * orr + oii * oii);
      float sn, cs;
      __sincosf(ph, &sn, &cs);
      v2f o;
      o.x = orr * cs - oii * sn;
      o.y = orr * sn + oii * cs;
      const long row = brow + wave * 16 + lhalf * 8 + v;  // batch index
      const long idx = (row << 6) + n * 16 + lmod;        // element index
      __builtin_nontemporal_store(o, (v2f*)(out + 2 * idx));
    }
  }
}

// ---------------------------------------------------------------------------
// Host-side launcher. Inputs: d_in[0]=x_real [B,64] f32, d_in[1]=x_imag,
// d_in[2]=phases [2016] f32. Output: complex64 [B,64] as interleaved floats.
// d_ws holds the composed unitary (planar Ur then Ui, 2*4096 floats = 32 KB).
// ---------------------------------------------------------------------------
extern "C" void kernel_launch(void* const* d_in, const int* in_sizes, int n_in,
                              void* d_out, int out_size, void* d_ws, size_t ws_size,
                              hipStream_t stream) {
  const float* x_real = (const float*)d_in[0];
  const float* x_imag = (const float*)d_in[1];
  const float* phases = (const float*)d_in[2];
  float* out = (float*)d_out;

  float* Ur = (float*)d_ws;
  float* Ui = Ur + 64 * 64;

  const int Bn = in_sizes[0] / 64;          // 262144
  const int nblocks = Bn / 64;              // 4096 (B is a multiple of 64)

  build_u_kernel<<<1, 64, 0, stream>>>(phases, Ur, Ui);
  photonic_kernel<<<nblocks, 128, SMEM_BYTES, stream>>>(x_real, x_imag, Ur, Ui, out);
}